// GraphSAGE_60902636257634
// MI455X (gfx1250) — compile-verified
//
#include <hip/hip_runtime.h>
#include <hip/hip_bf16.h>

typedef __attribute__((ext_vector_type(2))) float v2f;
typedef __attribute__((ext_vector_type(8))) float v8f;

#if defined(__HIP_DEVICE_COMPILE__) && !__has_builtin(__builtin_amdgcn_wmma_f32_16x16x4_f32)
#error "missing __builtin_amdgcn_wmma_f32_16x16x4_f32 on this toolchain"
#endif

#define NNODES 50000
#define NEDGES 800000
#define FIN 96

// ---------------------------------------------------------------------------
// Edge scatter: blockDim.x = 96 (one thread per feature, 3 waves), each block
// processes EPB edges. Gather feat[src] row (coalesced 384B), atomicAdd into
// agg[dst] row. Working set is L2-resident (192MB L2 >> 40MB), so atomics
// resolve in L2. cnt may be null (layer 2 reuses layer-1 counts).
// ---------------------------------------------------------------------------
__global__ void sage_scatter(const float* __restrict__ feat,
                             const long long* __restrict__ src,
                             const long long* __restrict__ dst,
                             float* __restrict__ agg,
                             float* __restrict__ cnt,
                             int n_edges, int epb) {
  const int f  = threadIdx.x;            // 0..95
  int e  = blockIdx.x * epb;
  int e1 = e + epb; if (e1 > n_edges) e1 = n_edges;
  for (; e < e1; ++e) {
    const int s = (int)src[e];           // uniform across block -> scalarizable
    const int d = (int)dst[e];
    atomicAdd(&agg[(size_t)d * FIN + f], feat[(size_t)s * FIN + f]);
    if (cnt != nullptr && f == 0) atomicAdd(&cnt[d], 1.0f);
  }
}

// ---------------------------------------------------------------------------
// Node transform via V_WMMA_F32_16X16X4_F32.
// out = xin @ Wl^T + bl + (agg/max(cnt,1)) @ Wr^T   (optionally ELU)
// One wave per 16-node tile; 3125 tiles exactly (no partial tiles, EXEC all-1s).
// A-frag: lane holds row m0+(lane&15), K pair at 4k + (lane>>4)*2.
// B-frag: lane holds col n0+(lane&15), same K pair (W row-major [F_OUT,96] is
//         exactly W^T column-major -> one float2 load per frag).
// B-fragments are preloaded into register ARRAYS so the 24 independent
// global_load_b64s pipeline ahead of the WMMA chain instead of forcing
// s_wait_loadcnt 0 before every WMMA (round-1 codegen pathology).
// ---------------------------------------------------------------------------
template <int F_OUT, bool DO_ELU>
__global__ void sage_node(const float* __restrict__ xin,
                          const float* __restrict__ agg,
                          const float* __restrict__ cnt,
                          const float* __restrict__ wl,   // [F_OUT, 96]
                          const float* __restrict__ bl,   // [F_OUT]
                          const float* __restrict__ wr,   // [F_OUT, 96]
                          float* __restrict__ out,
                          int n_tiles) {
  const int wave = threadIdx.x >> 5;
  const int lane = threadIdx.x & 31;
  const int tile = blockIdx.x * (blockDim.x >> 5) + wave;
  if (tile >= n_tiles) return;

  const int m0    = tile << 4;
  const int mrow  = m0 + (lane & 15);
  const int khalf = (lane >> 4) << 1;           // 0 or 2

  const float rc = 1.0f / fmaxf(cnt[mrow], 1.0f);

  // Preload all 24 K-chunks of A for both operands (self row + mean row).
  v2f ax[24], am[24];
  const float* xrow = xin + (size_t)mrow * FIN + khalf;
  const float* arow = agg + (size_t)mrow * FIN + khalf;
#pragma unroll
  for (int k = 0; k < 24; ++k) {
    ax[k] = *(const v2f*)(xrow + 4 * k);
    v2f t = *(const v2f*)(arow + 4 * k);
    am[k] = t * rc;
  }

#pragma unroll
  for (int nt = 0; nt < F_OUT / 16; ++nt) {
    const int col = (nt << 4) + (lane & 15);
    const float* wlcol = wl + (size_t)col * FIN + khalf;
    const float* wrcol = wr + (size_t)col * FIN + khalf;

    // Batch-load B fragments into register arrays (independent loads pipeline).
    v2f bl_[24], br_[24];
#pragma unroll
    for (int k = 0; k < 24; ++k) {
      bl_[k] = *(const v2f*)(wlcol + 4 * k);
      br_[k] = *(const v2f*)(wrcol + 4 * k);
    }

    v8f acc = {};
#pragma unroll
    for (int k = 0; k < 24; ++k)
      acc = __builtin_amdgcn_wmma_f32_16x16x4_f32(false, ax[k], false, bl_[k],
                                                  (short)0, acc, false, false);
#pragma unroll
    for (int k = 0; k < 24; ++k)
      acc = __builtin_amdgcn_wmma_f32_16x16x4_f32(false, am[k], false, br_[k],
                                                  (short)0, acc, false, false);

    const float bv = bl[col];
#pragma unroll
    for (int r = 0; r < 8; ++r) {
      float v = acc[r] + bv;
      if (DO_ELU) v = (v > 0.0f) ? v : (__expf(v) - 1.0f);
      const int row = m0 + r + ((lane >> 4) << 3);
      out[(size_t)row * F_OUT + col] = v;
    }
  }
}

extern "C" void kernel_launch(void* const* d_in, const int* in_sizes, int n_in,
                              void* d_out, int out_size, void* d_ws, size_t ws_size,
                              hipStream_t stream) {
  const float*     x   = (const float*)d_in[0];
  const long long* ei  = (const long long*)d_in[1];
  const float*     w1l = (const float*)d_in[2];
  const float*     b1l = (const float*)d_in[3];
  const float*     w1r = (const float*)d_in[4];
  const float*     w2l = (const float*)d_in[5];
  const float*     b2l = (const float*)d_in[6];
  const float*     w2r = (const float*)d_in[7];
  float*           out = (float*)d_out;

  const long long* src = ei;            // edge_index[0]
  const long long* dst = ei + NEDGES;   // edge_index[1]

  float* agg = (float*)d_ws;                       // N*96
  float* cnt = agg + (size_t)NNODES * FIN;         // N
  float* h   = cnt + NNODES;                       // N*96

  const int EPB         = 64;
  const int edge_blocks = (NEDGES + EPB - 1) / EPB;       // 12500
  const int TILES       = NNODES / 16;                    // 3125 (exact)
  const int WPB         = 8;                              // 256 threads = 8 waves
  const int node_blocks = (TILES + WPB - 1) / WPB;        // 391

  // ---- Layer 1 ----
  hipMemsetAsync(agg, 0, (size_t)NNODES * FIN * sizeof(float), stream);
  hipMemsetAsync(cnt, 0, (size_t)NNODES * sizeof(float), stream);
  sage_scatter<<<edge_blocks, 96, 0, stream>>>(x, src, dst, agg, cnt, NEDGES, EPB);
  sage_node<96, true><<<node_blocks, 32 * WPB, 0, stream>>>(
      x, agg, cnt, w1l, b1l, w1r, h, TILES);

  // ---- Layer 2 (counts unchanged; only re-zero agg) ----
  hipMemsetAsync(agg, 0, (size_t)NNODES * FIN * sizeof(float), stream);
  sage_scatter<<<edge_blocks, 96, 0, stream>>>(h, src, dst, agg, nullptr, NEDGES, EPB);
  sage_node<64, false><<<node_blocks, 32 * WPB, 0, stream>>>(
      h, agg, cnt, w2l, b2l, w2r, out, TILES);
}